// DiffusionLoss_44066364457449
// MI455X (gfx1250) — compile-verified
//
#include <hip/hip_runtime.h>
#include <math.h>

#define BATCH   2
#define N_TOKC  256
#define A_ATOMS 2048
#define EPSF    1e-12f

typedef float v2f __attribute__((ext_vector_type(2)));
typedef float v8f __attribute__((ext_vector_type(8)));

__device__ __forceinline__ float sigmoidf(float z) {
    return 1.0f / (1.0f + expf(-z));
}

// Block-wide sum for 256 threads (8 waves of 32). Result valid in thread 0.
__device__ __forceinline__ float block_sum_256(float v, float* scr) {
    #pragma unroll
    for (int o = 16; o > 0; o >>= 1) v += __shfl_xor(v, o, 32);
    int wid = threadIdx.x >> 5, ln = threadIdx.x & 31;
    __syncthreads();
    if (ln == 0) scr[wid] = v;
    __syncthreads();
    float r = (threadIdx.x < 8) ? scr[threadIdx.x] : 0.0f;
    if (wid == 0) {
        #pragma unroll
        for (int o = 4; o > 0; o >>= 1) r += __shfl_xor(r, o, 32);
    }
    return r;
}

// ---------------------------------------------------------------------------
// Kernel 1: per-batch setup. One block of 256 threads per batch.
// Computes token cumsum, per-atom features, squared norms, zeroes the pair
// accumulators, and reduces the 19 Kabsch/MSE moments.
// Moments: [0]=Swm [1]=Smask [2..4]=Sum wm*x [5..7]=Sum wm*gt
//          [8..16]=C[i][j]=Sum wm*x_i*gt_j (row-major) [17]=Sum wm|x|^2 [18]=Sum wm|gt|^2
// ---------------------------------------------------------------------------
__global__ __launch_bounds__(256)
void setup_kernel(const float* __restrict__ x, const float* __restrict__ gt,
                  const float* __restrict__ gtmask,
                  const float* __restrict__ isp, const float* __restrict__ isd,
                  const float* __restrict__ isr, const float* __restrict__ isl,
                  const float* __restrict__ tmask, const int* __restrict__ npt,
                  int* __restrict__ tokA, float* __restrict__ isl_a,
                  float* __restrict__ ispoly_a, float* __restrict__ isnuc_a,
                  float* __restrict__ nxA, float* __restrict__ ngtA,
                  float* __restrict__ mom, float* __restrict__ acc) {
    const int b = blockIdx.x;
    const int t = threadIdx.x;
    __shared__ int   s_cum[N_TOKC];
    __shared__ float s_isl[N_TOKC], s_isp[N_TOKC], s_inu[N_TOKC], s_w[N_TOKC];
    __shared__ float s_red[8];

    {
        const int gi = b * N_TOKC + t;
        const float tm = tmask[gi];
        const float p = isp[gi], d = isd[gi], r = isr[gi], l = isl[gi];
        s_isl[t] = l * tm;
        s_isp[t] = (p + d + r) * tm;
        s_inu[t] = (d + r) * tm;
        s_w[t]   = (1.0f + 5.0f * d + 5.0f * r + 10.0f * l) * tm;
        s_cum[t] = npt[gi];
    }
    __syncthreads();
    if (t == 0) {
        int run = 0;
        for (int i = 0; i < N_TOKC; i++) { run += s_cum[i]; s_cum[i] = run; }
    }
    __syncthreads();
    const int total = s_cum[N_TOKC - 1];

    float m[19];
    #pragma unroll
    for (int i = 0; i < 19; i++) m[i] = 0.0f;

    for (int k = 0; k < A_ATOMS / 256; k++) {
        const int l  = t + k * 256;
        const int gi = b * A_ATOMS + l;
        int tk = -1;
        if (l < total) {
            int lo = 0, hi = N_TOKC - 1;
            while (lo < hi) { int mid = (lo + hi) >> 1; if (s_cum[mid] > l) hi = mid; else lo = mid + 1; }
            tk = lo;
        }
        const float fisl = (tk >= 0) ? s_isl[tk] : 0.0f;
        const float fisp = (tk >= 0) ? s_isp[tk] : 0.0f;
        const float finu = (tk >= 0) ? s_inu[tk] : 0.0f;
        const float w    = (tk >= 0) ? s_w[tk]   : 0.0f;
        tokA[gi] = tk; isl_a[gi] = fisl; ispoly_a[gi] = fisp; isnuc_a[gi] = finu;

        const float xx = x[gi * 3 + 0], xy = x[gi * 3 + 1], xz = x[gi * 3 + 2];
        const float gx = gt[gi * 3 + 0], gy = gt[gi * 3 + 1], gz = gt[gi * 3 + 2];
        const float nx2 = xx * xx + xy * xy + xz * xz;
        const float ng2 = gx * gx + gy * gy + gz * gz;
        nxA[gi] = nx2; ngtA[gi] = ng2;

        const float msk = gtmask[gi];
        const float wm  = w * msk;
        m[0] += wm; m[1] += msk;
        m[2] += wm * xx; m[3] += wm * xy; m[4] += wm * xz;
        m[5] += wm * gx; m[6] += wm * gy; m[7] += wm * gz;
        m[8]  += wm * xx * gx; m[9]  += wm * xx * gy; m[10] += wm * xx * gz;
        m[11] += wm * xy * gx; m[12] += wm * xy * gy; m[13] += wm * xy * gz;
        m[14] += wm * xz * gx; m[15] += wm * xz * gy; m[16] += wm * xz * gz;
        m[17] += wm * nx2; m[18] += wm * ng2;
    }
    for (int i = 0; i < 19; i++) {
        const float r = block_sum_256(m[i], s_red);
        if (t == 0) mom[b * 19 + i] = r;
    }
    if (t < 5) acc[b * 5 + t] = 0.0f;
}

// ---------------------------------------------------------------------------
// Kernel 2: pairwise stage. 128 threads = 4 waves; each wave owns one 16x16
// (i,j) tile; block covers 4 consecutive j tiles at one i tile.
// Gram tiles of x and gt computed with V_WMMA_F32_16X16X4_F32 (K=3 padded).
// Accumulators per batch: [0]=bond_num [1]=bond_den [2]=ce [3]=c [4]=m2sum
// ---------------------------------------------------------------------------
__global__ __launch_bounds__(128)
void pair_kernel(const float* __restrict__ x, const float* __restrict__ gt,
                 const float* __restrict__ gtmask, const float* __restrict__ tb,
                 const int* __restrict__ tokA, const float* __restrict__ isl_a,
                 const float* __restrict__ ispoly_a, const float* __restrict__ isnuc_a,
                 const float* __restrict__ nxA, const float* __restrict__ ngtA,
                 float* __restrict__ acc) {
    const int lane = threadIdx.x & 31;
    const int wv   = threadIdx.x >> 5;
    int bi = blockIdx.x;
    const int jg = bi & 31;  bi >>= 5;
    const int it = bi & 127; bi >>= 7;
    const int b  = bi;
    const int i0 = it * 16;
    const int j0 = (jg * 4 + wv) * 16;
    const int base = b * A_ATOMS;

    __shared__ float r_nx[16], r_ng[16], r_mk[16], r_nu[16], r_il[16];
    __shared__ int   r_tk[16];
    __shared__ float part[5][4];
    if (threadIdx.x < 16) {
        const int gi = base + i0 + threadIdx.x;
        r_nx[threadIdx.x] = nxA[gi];
        r_ng[threadIdx.x] = ngtA[gi];
        r_mk[threadIdx.x] = gtmask[gi];
        r_nu[threadIdx.x] = isnuc_a[gi];
        r_il[threadIdx.x] = isl_a[gi];
        r_tk[threadIdx.x] = tokA[gi];
    }
    __syncthreads();

    // Fragments: A (rows i0..i0+15) and B (cols j0..j0+15), K = {x,y,z,0}.
    // Layout (f32 16x16x4): lanes 0-15 hold K=0 (v.x)/K=1 (v.y); lanes 16-31 K=2/K=3.
    const int la = lane & 15;
    const float* xr = x  + (size_t)(base + i0 + la) * 3;
    const float* xc = x  + (size_t)(base + j0 + la) * 3;
    const float* gr = gt + (size_t)(base + i0 + la) * 3;
    const float* gc = gt + (size_t)(base + j0 + la) * 3;
    v2f ax, bx, ag, bg;
    if (lane < 16) {
        ax.x = xr[0]; ax.y = xr[1];  bx.x = xc[0]; bx.y = xc[1];
        ag.x = gr[0]; ag.y = gr[1];  bg.x = gc[0]; bg.y = gc[1];
    } else {
        ax.x = xr[2]; ax.y = 0.0f;   bx.x = xc[2]; bx.y = 0.0f;
        ag.x = gr[2]; ag.y = 0.0f;   bg.x = gc[2]; bg.y = 0.0f;
    }

    v8f cx = {};
    v8f cg = {};
    cx = __builtin_amdgcn_wmma_f32_16x16x4_f32(false, ax, false, bx, (short)0, cx, false, false);
    cg = __builtin_amdgcn_wmma_f32_16x16x4_f32(false, ag, false, bg, (short)0, cg, false, false);

    // Column (j) data: fixed per lane.
    const int jn = j0 + la;
    const float nxj = nxA[base + jn], ngj = ngtA[base + jn];
    const float mkj = gtmask[base + jn];
    const float ipj = ispoly_a[base + jn];
    int tkj = tokA[base + jn]; if (tkj < 0) tkj = 0;
    const float* tbb = tb + (size_t)b * N_TOKC * N_TOKC;

    const int ihalf = (lane < 16) ? 0 : 8;
    float s_bn = 0.0f, s_bd = 0.0f, s_ce = 0.0f, s_cs = 0.0f, s_ms = 0.0f;
    #pragma unroll
    for (int r = 0; r < 8; r++) {
        const int il = ihalf + r;       // local row 0..15  (C VGPR r, lane half)
        const int im = i0 + il;         // global row atom index
        const float d2x = r_nx[il] + nxj - 2.0f * cx[r];
        const float d2g = r_ng[il] + ngj - 2.0f * cg[r];
        const float dxv = sqrtf(fmaxf(d2x, 0.0f) + EPSF);
        const float dgv = sqrtf(fmaxf(d2g, 0.0f) + EPSF);
        const float pm   = r_mk[il] * mkj;
        const float diff = dgv - dxv;
        const float d    = fabsf(diff);
        const float e = 0.25f * (sigmoidf(0.5f - d) + sigmoidf(1.0f - d) +
                                 sigmoidf(2.0f - d) + sigmoidf(4.0f - d));
        const float inu  = r_nu[il];
        const float ccut = ((dgv < 30.0f) ? inu : 0.0f) +
                           ((dgv < 15.0f) ? (1.0f - inu) : 0.0f);
        const float m2   = (im == jn) ? 0.0f : pm;
        int tki = r_tk[il]; if (tki < 0) tki = 0;
        const float mb = tbb[tki * N_TOKC + tkj] * r_il[il] * ipj * pm;
        s_bn += diff * diff * mb;
        s_bd += mb;
        s_ce += ccut * e * m2;
        s_cs += ccut * m2;
        s_ms += m2;
    }
    #pragma unroll
    for (int o = 16; o > 0; o >>= 1) {
        s_bn += __shfl_xor(s_bn, o, 32);
        s_bd += __shfl_xor(s_bd, o, 32);
        s_ce += __shfl_xor(s_ce, o, 32);
        s_cs += __shfl_xor(s_cs, o, 32);
        s_ms += __shfl_xor(s_ms, o, 32);
    }
    if (lane == 0) {
        part[0][wv] = s_bn; part[1][wv] = s_bd; part[2][wv] = s_ce;
        part[3][wv] = s_cs; part[4][wv] = s_ms;
    }
    __syncthreads();
    if (threadIdx.x < 5) {
        const float v = part[threadIdx.x][0] + part[threadIdx.x][1] +
                        part[threadIdx.x][2] + part[threadIdx.x][3];
        atomicAdd(&acc[b * 5 + threadIdx.x], v);
    }
}

// ---------------------------------------------------------------------------
// Kernel 3: finish. Horn quaternion Kabsch (power iteration on 4x4 N),
// closed-form weighted MSE from moments, per-batch loss, batch mean.
// ---------------------------------------------------------------------------
__global__ __launch_bounds__(64)
void finish_kernel(const float* __restrict__ tvec, const float* __restrict__ mom,
                   const float* __restrict__ acc, float* __restrict__ out) {
    __shared__ float losses[BATCH];
    const int b = threadIdx.x;
    if (b < BATCH) {
        const float* m = mom + b * 19;
        const float Swm = m[0], Smask = m[1];
        const float inv = 1.0f / Swm;
        float muB[3] = {m[2] * inv, m[3] * inv, m[4] * inv};  // x centroid
        float muA[3] = {m[5] * inv, m[6] * inv, m[7] * inv};  // gt centroid
        float H[3][3];                                        // H[i][j]=Sum wm*(x-muB)_i*(gt-muA)_j
        #pragma unroll
        for (int i = 0; i < 3; i++)
            #pragma unroll
            for (int j = 0; j < 3; j++)
                H[i][j] = m[8 + i * 3 + j] - Swm * muB[i] * muA[j];

        // Horn: S_ab = sum s_a t_b with s=gt-centered (source), t=x-centered.
        const float Sxx = H[0][0], Syy = H[1][1], Szz = H[2][2];
        const float Sxy = H[1][0], Sxz = H[2][0], Syx = H[0][1];
        const float Syz = H[2][1], Szx = H[0][2], Szy = H[1][2];
        const float N00 = Sxx + Syy + Szz;
        const float N01 = Syz - Szy, N02 = Szx - Sxz, N03 = Sxy - Syx;
        const float N11 = Sxx - Syy - Szz;
        const float N12 = Sxy + Syx, N13 = Szx + Sxz;
        const float N22 = -Sxx + Syy - Szz;
        const float N23 = Syz + Szy;
        const float N33 = -Sxx - Syy + Szz;
        const float sh = 1.0f + 2.0f * (fabsf(Sxx) + fabsf(Syy) + fabsf(Szz) +
                                        fabsf(Sxy) + fabsf(Syx) + fabsf(Sxz) +
                                        fabsf(Szx) + fabsf(Syz) + fabsf(Szy));
        float q0 = 1.0f, q1 = 0.0123f, q2 = 0.0234f, q3 = 0.0345f;
        for (int it = 0; it < 128; it++) {
            const float t0 = N00 * q0 + N01 * q1 + N02 * q2 + N03 * q3 + sh * q0;
            const float t1 = N01 * q0 + N11 * q1 + N12 * q2 + N13 * q3 + sh * q1;
            const float t2 = N02 * q0 + N12 * q1 + N22 * q2 + N23 * q3 + sh * q2;
            const float t3 = N03 * q0 + N13 * q1 + N23 * q2 + N33 * q3 + sh * q3;
            const float n = rsqrtf(t0 * t0 + t1 * t1 + t2 * t2 + t3 * t3 + 1e-30f);
            q0 = t0 * n; q1 = t1 * n; q2 = t2 * n; q3 = t3 * n;
        }
        float R[3][3];
        R[0][0] = 1.0f - 2.0f * (q2 * q2 + q3 * q3);
        R[0][1] = 2.0f * (q1 * q2 - q0 * q3);
        R[0][2] = 2.0f * (q1 * q3 + q0 * q2);
        R[1][0] = 2.0f * (q1 * q2 + q0 * q3);
        R[1][1] = 1.0f - 2.0f * (q1 * q1 + q3 * q3);
        R[1][2] = 2.0f * (q2 * q3 - q0 * q1);
        R[2][0] = 2.0f * (q1 * q3 - q0 * q2);
        R[2][1] = 2.0f * (q2 * q3 + q0 * q1);
        R[2][2] = 1.0f - 2.0f * (q1 * q1 + q2 * q2);
        float tr = 0.0f;
        #pragma unroll
        for (int i = 0; i < 3; i++)
            #pragma unroll
            for (int j = 0; j < 3; j++)
                tr += R[i][j] * H[i][j];

        const float TtT = m[17] - Swm * (muB[0]*muB[0] + muB[1]*muB[1] + muB[2]*muB[2]);
        const float SsS = m[18] - Swm * (muA[0]*muA[0] + muA[1]*muA[1] + muA[2]*muA[2]);
        const float l_mse = (TtT + SsS - 2.0f * tr) / (3.0f * Smask);

        const float* a = acc + b * 5;
        const float l_bond = a[0] / a[1];
        const float l_lddt = 1.0f - a[2] / a[3];
        const float tb_ = tvec[b];
        const float w_t = (tb_ * tb_ + 256.0f) / ((tb_ + 16.0f) * (tb_ + 16.0f));
        losses[b] = w_t * (l_mse + l_bond) + l_lddt;
    }
    __syncthreads();
    if (threadIdx.x == 0) {
        float s = 0.0f;
        for (int i = 0; i < BATCH; i++) s += losses[i];
        out[0] = s / (float)BATCH;
    }
}

extern "C" void kernel_launch(void* const* d_in, const int* in_sizes, int n_in,
                              void* d_out, int out_size, void* d_ws, size_t ws_size,
                              hipStream_t stream) {
    const float* x     = (const float*)d_in[0];
    const float* gt    = (const float*)d_in[1];
    const float* gmask = (const float*)d_in[2];
    const float* isp   = (const float*)d_in[3];
    const float* isd   = (const float*)d_in[4];
    const float* isr   = (const float*)d_in[5];
    const float* isl   = (const float*)d_in[6];
    const float* tb    = (const float*)d_in[7];
    const float* tmask = (const float*)d_in[8];
    const int*   npt   = (const int*)d_in[9];
    const float* tvec  = (const float*)d_in[10];

    const int BA = BATCH * A_ATOMS;
    int*   tokp     = (int*)d_ws;
    float* f        = (float*)d_ws;
    float* isl_a    = f + BA;
    float* ispoly_a = isl_a + BA;
    float* isnuc_a  = ispoly_a + BA;
    float* nxA      = isnuc_a + BA;
    float* ngtA     = nxA + BA;
    float* mom      = ngtA + BA;          // BATCH*19
    float* accum    = mom + BATCH * 19;   // BATCH*5

    setup_kernel<<<BATCH, 256, 0, stream>>>(x, gt, gmask, isp, isd, isr, isl, tmask,
                                            npt, tokp, isl_a, ispoly_a, isnuc_a,
                                            nxA, ngtA, mom, accum);
    pair_kernel<<<BATCH * 128 * 32, 128, 0, stream>>>(x, gt, gmask, tb, tokp, isl_a,
                                                      ispoly_a, isnuc_a, nxA, ngtA,
                                                      accum);
    finish_kernel<<<1, 64, 0, stream>>>(tvec, mom, accum, (float*)d_out);
}